// GAE_29351806501366
// MI455X (gfx1250) — compile-verified
//
#include <hip/hip_runtime.h>
#include <math.h>

typedef __attribute__((ext_vector_type(2))) float v2f;
typedef __attribute__((ext_vector_type(8))) float v8f;

// ---------------- degree / normalization ----------------
__global__ void k_deg_init(float* deg, int N) {
    int i = blockIdx.x * blockDim.x + threadIdx.x;
    if (i < N) deg[i] = 1.0f;  // self-loop contributes 1
}

__global__ void k_deg_acc(const int* __restrict__ dst, float* deg, int E) {
    int i = blockIdx.x * blockDim.x + threadIdx.x;
    if (i < E) atomicAdd(&deg[dst[i]], 1.0f);
}

__global__ void k_rsqrt_inplace(float* deg, int N) {
    int i = blockIdx.x * blockDim.x + threadIdx.x;
    if (i < N) deg[i] = rsqrtf(deg[i]);
}

// ---------------- WMMA f32 GEMM: C[M,N] = A[M,K] @ B[K,N] (all row-major) ----
// M % 16 == 0, N % 16 == 0, K % 4 == 0. One wave computes one 16x16 C tile.
__global__ void k_wmma_gemm(const float* __restrict__ A, const float* __restrict__ B,
                            float* __restrict__ C, int M, int N, int K) {
    int wave   = (blockIdx.x * blockDim.x + threadIdx.x) >> 5;
    int tilesN = N >> 4;
    int tiles  = (M >> 4) * tilesN;
    if (wave >= tiles) return;            // wave-uniform branch: EXEC stays all-ones
    int tm = wave / tilesN;
    int tn = wave % tilesN;

    int lane = threadIdx.x & 31;
    int half = lane >> 4;                 // 0: K pair {0,1}; 1: K pair {2,3}
    int r    = lane & 15;
    int kb   = half << 1;

    const float* Arow = A + (size_t)(tm * 16 + r) * K;   // A: row r, contiguous K pair
    const float* Bcol = B + (tn * 16 + r);               // B: col r, N-strided K pair

    v8f acc = {};
    for (int k0 = 0; k0 < K; k0 += 4) {
        v2f a, b;
        a.x = Arow[k0 + kb];
        a.y = Arow[k0 + kb + 1];
        b.x = Bcol[(size_t)(k0 + kb) * N];
        b.y = Bcol[(size_t)(k0 + kb + 1) * N];
        acc = __builtin_amdgcn_wmma_f32_16x16x4_f32(false, a, false, b,
                                                    (short)0, acc, false, false);
    }

    int col = tn * 16 + r;
    float* Crow = C + (size_t)(tm * 16 + half * 8) * N + col;
#pragma unroll
    for (int i = 0; i < 8; ++i) Crow[(size_t)i * N] = acc[i];
}

// ---------------- GCN aggregation ----------------
// out[i,f] = hlin[i,f] * dis[i]^2 + bias[f]   (self-loop term + bias)
__global__ void k_agg_init(const float* __restrict__ hlin, const float* __restrict__ dis,
                           const float* __restrict__ bias, float* __restrict__ out,
                           int total, int fshift, int fmask) {
    int idx = blockIdx.x * blockDim.x + threadIdx.x;
    if (idx >= total) return;
    int i = idx >> fshift;
    int f = idx & fmask;
    float di = dis[i];
    out[idx] = hlin[idx] * di * di + bias[f];
}

// scatter: out[dst,f] += hlin[src,f] * dis[src]*dis[dst]; blockDim.x == F
__global__ void k_agg_edge(const int* __restrict__ src, const int* __restrict__ dst,
                           const float* __restrict__ dis, const float* __restrict__ hlin,
                           float* __restrict__ out, int F) {
    int e = blockIdx.x;
    int f = threadIdx.x;
    int s = src[e];
    int d = dst[e];
    float norm = dis[s] * dis[d];
    atomicAdd(&out[(size_t)d * F + f], hlin[(size_t)s * F + f] * norm);
}

__global__ void k_lrelu(float* x, int n) {
    int i = blockIdx.x * blockDim.x + threadIdx.x;
    if (i >= n) return;
    float v = x[i];
    x[i] = v > 0.0f ? v : 0.01f * v;
}

// ---------------- decode: out = sigmoid(H @ H^T), H: [Nn,K] row-major --------
// B tile = rows of H  ->  identical contiguous float2 load pattern as A tile.
__global__ void k_decode(const float* __restrict__ H, float* __restrict__ out,
                         int Nn, int K) {
    int wave   = (blockIdx.x * blockDim.x + threadIdx.x) >> 5;
    int tilesN = Nn >> 4;
    int tiles  = tilesN * tilesN;
    if (wave >= tiles) return;
    int tm = wave / tilesN;
    int tn = wave % tilesN;

    int lane = threadIdx.x & 31;
    int half = lane >> 4;
    int r    = lane & 15;
    int kb   = half << 1;

    const float* Ar = H + (size_t)(tm * 16 + r) * K;
    const float* Br = H + (size_t)(tn * 16 + r) * K;

    v8f acc = {};
#pragma unroll 8
    for (int k0 = 0; k0 < K; k0 += 4) {
        v2f a, b;
        a.x = Ar[k0 + kb];
        a.y = Ar[k0 + kb + 1];
        b.x = Br[k0 + kb];
        b.y = Br[k0 + kb + 1];
        acc = __builtin_amdgcn_wmma_f32_16x16x4_f32(false, a, false, b,
                                                    (short)0, acc, false, false);
    }

    int col = tn * 16 + r;
    float* Orow = out + (size_t)(tm * 16 + half * 8) * Nn + col;
#pragma unroll
    for (int i = 0; i < 8; ++i) {
        float v = acc[i];
        Orow[(size_t)i * Nn] = 1.0f / (1.0f + __expf(-v));
    }
}

static inline int ilog2(int v) { int s = 0; while ((1 << s) < v) ++s; return s; }

extern "C" void kernel_launch(void* const* d_in, const int* in_sizes, int n_in,
                              void* d_out, int out_size, void* d_ws, size_t ws_size,
                              hipStream_t stream) {
    const float* x  = (const float*)d_in[0];
    const int*   ei = (const int*)  d_in[1];
    const float* W1 = (const float*)d_in[3];
    const float* b1 = (const float*)d_in[4];
    const float* W2 = (const float*)d_in[5];
    const float* b2 = (const float*)d_in[6];

    const int N   = in_sizes[2];          // batch has one entry per node
    const int E   = in_sizes[1] / 2;
    const int Fin = in_sizes[0] / N;      // 512
    const int F1  = in_sizes[4];          // 256
    const int F2  = in_sizes[6];          // 128

    const int* src = ei;                  // edge_index[0]
    const int* dst = ei + E;              // edge_index[1]

    // workspace layout (floats)
    float* degdis = (float*)d_ws;                 // N   (degree, then rsqrt in place)
    float* h1lin  = degdis + N;                   // N*F1
    float* a1     = h1lin + (size_t)N * F1;       // N*F1 (aggregated layer 1 -> h1)
    float* h2lin  = a1 + (size_t)N * F1;          // N*F2

    float* out  = (float*)d_out;                  // x1: N*N
    float* hout = out + (size_t)N * N;            // h:  N*F2

    const int TB = 256;

    // 1) degrees with self-loops, then dis = deg^-1/2
    k_deg_init<<<(N + TB - 1) / TB, TB, 0, stream>>>(degdis, N);
    k_deg_acc<<<(E + TB - 1) / TB, TB, 0, stream>>>(dst, degdis, E);
    k_rsqrt_inplace<<<(N + TB - 1) / TB, TB, 0, stream>>>(degdis, N);

    // 2) layer 1: h1lin = x @ W1
    {
        int tiles  = (N / 16) * (F1 / 16);
        int blocks = (tiles + 7) / 8;             // 8 waves per 256-thread block
        k_wmma_gemm<<<blocks, TB, 0, stream>>>(x, W1, h1lin, N, F1, Fin);
    }
    // 3) aggregate + bias, scatter edges, leaky relu
    {
        int total = N * F1;
        k_agg_init<<<(total + TB - 1) / TB, TB, 0, stream>>>(h1lin, degdis, b1, a1,
                                                             total, ilog2(F1), F1 - 1);
        k_agg_edge<<<E, F1, 0, stream>>>(src, dst, degdis, h1lin, a1, F1);
        k_lrelu<<<(total + TB - 1) / TB, TB, 0, stream>>>(a1, total);
    }

    // 4) layer 2: h2lin = h1 @ W2
    {
        int tiles  = (N / 16) * (F2 / 16);
        int blocks = (tiles + 7) / 8;
        k_wmma_gemm<<<blocks, TB, 0, stream>>>(a1, W2, h2lin, N, F2, F1);
    }
    // 5) aggregate layer 2 directly into h output slot, leaky relu in place
    {
        int total = N * F2;
        k_agg_init<<<(total + TB - 1) / TB, TB, 0, stream>>>(h2lin, degdis, b2, hout,
                                                             total, ilog2(F2), F2 - 1);
        k_agg_edge<<<E, F2, 0, stream>>>(src, dst, degdis, h2lin, hout, F2);
        k_lrelu<<<(total + TB - 1) / TB, TB, 0, stream>>>(hout, total);
    }

    // 6) decode: x1 = sigmoid(h @ h^T)  (store-bandwidth-bound: 400 MB of f32)
    {
        int tiles  = (N / 16) * (N / 16);
        int blocks = (tiles + 7) / 8;
        k_decode<<<blocks, TB, 0, stream>>>(hout, out, N, F2);
    }
}